// ChatbotModel_10436770530019
// MI455X (gfx1250) — compile-verified
//
#include <hip/hip_runtime.h>
#include <hip/hip_bf16.h>

#define SEQL  128
#define BATCH 32
#define HID   512
#define G4    2048          // 4*HID gate dim
#define OUTD  32000
#define LSTM_WGS 16         // persistent workgroups for the scan
#define GS_STR 129          // padded stride for gate-exchange tile

typedef float v2f __attribute__((ext_vector_type(2)));
typedef float v8f __attribute__((ext_vector_type(8)));
typedef int   v4i __attribute__((ext_vector_type(4)));

// Pointer types matching the async-builtin prototype (from hipcc diagnostic):
// arg0 = global (AS1) v4i*, arg1 = LDS (AS3) v4i*, then imm offset, imm cpol.
typedef __attribute__((address_space(1))) v4i* gbl_v4i_p;
typedef __attribute__((address_space(3))) v4i* lds_v4i_p;

// ---- CDNA5 async global->LDS path (guarded; falls back to sync copy) ------
#if defined(__has_builtin)
#if __has_builtin(__builtin_amdgcn_global_load_async_to_lds_b128)
#define ASYNC_LDS_COPY 1
#endif
#endif
#ifndef ASYNC_LDS_COPY
#define ASYNC_LDS_COPY 0
#endif

#if ASYNC_LDS_COPY
__device__ __forceinline__ void wait_async0() {
#if __has_builtin(__builtin_amdgcn_s_wait_asynccnt)
  __builtin_amdgcn_s_wait_asynccnt(0);
#else
  asm volatile("s_wait_asynccnt 0x0" ::: "memory");
#endif
}
#endif

// D = A(16x4 f32) * B(4x16 f32) + C(16x16 f32), full fp32 matrix pipe.
__device__ __forceinline__ v8f wmma4(v2f a, v2f b, v8f c) {
  return __builtin_amdgcn_wmma_f32_16x16x4_f32(false, a, false, b, (short)0, c,
                                               false, false);
}

__device__ __forceinline__ float sigmoidf_(float x) {
  return 1.0f / (1.0f + __builtin_expf(-x));
}

// ---------------------------------------------------------------------------
// Zero h ping-pong buffer and per-step barrier counters (runs every call).
__global__ void init_kernel(float* hbuf, int* bar) {
  int i = blockIdx.x * 256 + threadIdx.x;
  if (i < 2 * BATCH * HID) hbuf[i] = 0.0f;
  if (i < SEQL) bar[i] = 0;
}

// ---------------------------------------------------------------------------
// x_gates = gather(emb_table, x) @ W_ih^T + (b_ih + b_hh)
// M = 4096 (s*B+b), N = 2048, K = 512.  WG tile 64x128, 8 waves x 4 C-tiles.
__global__ __launch_bounds__(256) void embed_gates_kernel(
    const int* __restrict__ x, const float* __restrict__ emb,
    const float* __restrict__ W_ih, const float* __restrict__ b_ih,
    const float* __restrict__ b_hh, float* __restrict__ xgout) {
  const int nbase = blockIdx.x * 128;
  const int mbase = blockIdx.y * 64;
  const int tid = threadIdx.x, lane = tid & 31, w = tid >> 5;
  const int mt = w & 3, ntg = w >> 2;

  const int am = mbase + mt * 16 + (lane & 15);
  const int row = x[am];  // embedding gather (A-matrix row)
  const float2* arow = (const float2*)(emb + (size_t)row * HID);
  const int kaoff = (lane >> 4) << 1;  // A: lanes16-31 hold K+2,K+3
  const int kboff = lane >> 4;         // B: lanes16-31 hold next K row

  v8f acc[4] = {};
  for (int k0 = 0; k0 < HID; k0 += 4) {
    float2 av = arow[(k0 + kaoff) >> 1];
    v2f a; a[0] = av.x; a[1] = av.y;
    const int kb = k0 + kboff;
#pragma unroll
    for (int i = 0; i < 4; ++i) {
      const int n = nbase + (ntg * 4 + i) * 16 + (lane & 15);
      const float* brow = W_ih + (size_t)n * HID;
      v2f b; b[0] = brow[kb]; b[1] = brow[kb + 2];
      acc[i] = wmma4(a, b, acc[i]);
    }
  }
  const int mrow = mbase + mt * 16 + ((lane >> 4) << 3);
#pragma unroll
  for (int i = 0; i < 4; ++i) {
    const int n = nbase + (ntg * 4 + i) * 16 + (lane & 15);
    const float bias = b_ih[n] + b_hh[n];
#pragma unroll
    for (int r = 0; r < 8; ++r)
      xgout[(size_t)(mrow + r) * G4 + n] = acc[i][r] + bias;
  }
}

// ---------------------------------------------------------------------------
// Persistent LSTM scan: 16 WGs, each owns 32 hidden units (128 W_hh rows
// = 256 KB f32, resident in CDNA5's 320 KB LDS for all 128 steps).
// h ping-pongs in global scratch in k-major [512][32] layout and is staged
// into LDS each step through the async global->LDS DMA path (ASYNCcnt).
// Per-step device-scope atomic barrier synchronizes the 16 WGs.
__global__ __launch_bounds__(256) void lstm_scan_kernel(
    const float* __restrict__ xg, const float* __restrict__ W_hh,
    float* __restrict__ hbuf, float* __restrict__ hs, int* __restrict__ bar) {
  extern __shared__ float lds[];
  float* Bs  = lds;               // [512][128]: W_hh slice, transposed (256 KB)
  float* HsT = lds + HID * 128;   // [512][32] : h stage, k-major (64 KB)
  float* Gs  = HsT;               // [32][GS_STR]: gate exchange (aliased)

  const int j0 = blockIdx.x * 32;  // this WG's hidden-unit slice
  const int tid = threadIdx.x, lane = tid & 31, w = tid >> 5;
  const int mt = w & 1, ntp = w >> 1;  // 2 M-tiles x (4 wave-pairs x 2 N-tiles)

  // Preload: Bs[k*128+n] = W_hh[(n/32)*512 + j0 + n%32][k]  (coalesced reads)
  for (int nr = 0; nr < 128; ++nr) {
    const int grow = (nr >> 5) * HID + j0 + (nr & 31);
    const float* src = W_hh + (size_t)grow * HID;
    for (int k = tid; k < HID; k += 256) Bs[k * 128 + nr] = src[k];
  }

  const int am = mt * 16 + (lane & 15);     // batch row for A fragment
  const int kaoff = (lane >> 4) << 1;
  const int kboff = lane >> 4;
  const int n0 = ntp * 32 + (lane & 15);

  const int ejl = tid >> 3;         // elementwise: local hidden j 0..31
  const int ebq = (tid & 7) * 4;    // 4 consecutive batches per thread
  float creg[4] = {0.f, 0.f, 0.f, 0.f};  // cell state lives in registers

  __syncthreads();

  for (int s = 0; s < SEQL; ++s) {
    const float* hin = hbuf + (s & 1) * (BATCH * HID);
    float* hout = hbuf + ((s + 1) & 1) * (BATCH * HID);

    // Stage h[512][32] into LDS: 4096 16B chunks, async DMA when available.
    {
      const float4* src4 = (const float4*)hin;
      float4* dst4 = (float4*)HsT;
#pragma unroll
      for (int i = 0; i < 16; ++i) {
        const int c = tid + i * 256;
#if ASYNC_LDS_COPY
        __builtin_amdgcn_global_load_async_to_lds_b128(
            (gbl_v4i_p)(src4 + c), (lds_v4i_p)(dst4 + c), 0, 0);
#else
        dst4[c] = src4[c];
#endif
      }
#if ASYNC_LDS_COPY
      wait_async0();
#endif
    }
    __syncthreads();

    // gates_slice[32,128] = h_prev[32,512] @ Bs[512,128]
    v8f acc0 = {}, acc1 = {};
    for (int k0 = 0; k0 < HID; k0 += 4) {
      const int ka = k0 + kaoff;
      v2f a; a[0] = HsT[ka * 32 + am]; a[1] = HsT[(ka + 1) * 32 + am];
      const int kb = k0 + kboff;
      v2f b0, b1;
      b0[0] = Bs[kb * 128 + n0];        b0[1] = Bs[(kb + 2) * 128 + n0];
      b1[0] = Bs[kb * 128 + n0 + 16];   b1[1] = Bs[(kb + 2) * 128 + n0 + 16];
      acc0 = wmma4(a, b0, acc0);
      acc1 = wmma4(a, b1, acc1);
    }
    __syncthreads();  // all waves done reading HsT before Gs overwrites it

    // exchange gate tiles through LDS (i,f,g,o for one j live in 4 waves)
    const int mrow = mt * 16 + ((lane >> 4) << 3);
#pragma unroll
    for (int r = 0; r < 8; ++r) {
      Gs[(mrow + r) * GS_STR + n0]      = acc0[r];
      Gs[(mrow + r) * GS_STR + n0 + 16] = acc1[r];
    }
    __syncthreads();

    // elementwise LSTM cell update for (j = j0+ejl, batches ebq..ebq+3)
    const int jg = j0 + ejl;
    float hvals[4];
#pragma unroll
    for (int q = 0; q < 4; ++q) {
      const int b = ebq + q;
      const float* xr = xg + (size_t)(s * BATCH + b) * G4;
      float gi = Gs[b * GS_STR + ejl]       + xr[jg];
      float gf = Gs[b * GS_STR + 32 + ejl]  + xr[HID + jg];
      float gg = Gs[b * GS_STR + 64 + ejl]  + xr[2 * HID + jg];
      float go = Gs[b * GS_STR + 96 + ejl]  + xr[3 * HID + jg];
      gi = sigmoidf_(gi); gf = sigmoidf_(gf);
      gg = tanhf(gg);     go = sigmoidf_(go);
      const float cn = gf * creg[q] + gi * gg;
      creg[q] = cn;
      hvals[q] = go * tanhf(cn);
      hs[(size_t)(s * BATCH + b) * HID + jg] = hvals[q];  // seq output (row-major)
    }
    // k-major h store: 4 consecutive batches -> one coalesced float4
    *(float4*)(hout + jg * 32 + ebq) =
        make_float4(hvals[0], hvals[1], hvals[2], hvals[3]);

    // grid-wide step barrier: release our h slice, acquire everyone else's
    __threadfence();
    __syncthreads();
    if (tid == 0)
      __hip_atomic_fetch_add(&bar[s], 1, __ATOMIC_RELEASE,
                             __HIP_MEMORY_SCOPE_AGENT);
    while (__hip_atomic_load(&bar[s], __ATOMIC_ACQUIRE,
                             __HIP_MEMORY_SCOPE_AGENT) < LSTM_WGS) {
      __builtin_amdgcn_s_sleep(2);
    }
    __syncthreads();
  }
}

// ---------------------------------------------------------------------------
// logits = hs[4096,512] @ W_fc^T[512,32000] + b_fc.  Bound by the 524 MB
// output write; W_fc stays L2-resident.  WG tile 64x128.
__global__ __launch_bounds__(256) void fc_kernel(
    const float* __restrict__ A, const float* __restrict__ Wf,
    const float* __restrict__ bf, float* __restrict__ out) {
  const int nbase = blockIdx.x * 128;
  const int mbase = blockIdx.y * 64;
  const int tid = threadIdx.x, lane = tid & 31, w = tid >> 5;
  const int mt = w & 3, ntg = w >> 2;

  const int am = mbase + mt * 16 + (lane & 15);
  const float2* arow = (const float2*)(A + (size_t)am * HID);
  const int kaoff = (lane >> 4) << 1;
  const int kboff = lane >> 4;

  v8f acc[4] = {};
  for (int k0 = 0; k0 < HID; k0 += 4) {
    float2 av = arow[(k0 + kaoff) >> 1];
    v2f a; a[0] = av.x; a[1] = av.y;
    const int kb = k0 + kboff;
#pragma unroll
    for (int i = 0; i < 4; ++i) {
      const int n = nbase + (ntg * 4 + i) * 16 + (lane & 15);
      const float* brow = Wf + (size_t)n * HID;
      v2f b; b[0] = brow[kb]; b[1] = brow[kb + 2];
      acc[i] = wmma4(a, b, acc[i]);
    }
  }
  const int mrow = mbase + mt * 16 + ((lane >> 4) << 3);
#pragma unroll
  for (int i = 0; i < 4; ++i) {
    const int n = nbase + (ntg * 4 + i) * 16 + (lane & 15);
    const float bias = bf[n];
#pragma unroll
    for (int r = 0; r < 8; ++r)
      out[(size_t)(mrow + r) * OUTD + n] = acc[i][r] + bias;
  }
}

// ---------------------------------------------------------------------------
extern "C" void kernel_launch(void* const* d_in, const int* in_sizes, int n_in,
                              void* d_out, int out_size, void* d_ws,
                              size_t ws_size, hipStream_t stream) {
  const int*   x    = (const int*)d_in[0];
  const float* emb  = (const float*)d_in[1];
  const float* W_ih = (const float*)d_in[2];
  const float* W_hh = (const float*)d_in[3];
  const float* b_ih = (const float*)d_in[4];
  const float* b_hh = (const float*)d_in[5];
  const float* W_fc = (const float*)d_in[6];
  const float* b_fc = (const float*)d_in[7];
  float* out = (float*)d_out;

  // workspace layout (~42 MB)
  float* xg    = (float*)d_ws;                          // [4096][2048]
  float* hsbuf = xg + (size_t)SEQL * BATCH * G4;        // [4096][512]
  float* hbuf  = hsbuf + (size_t)SEQL * BATCH * HID;    // 2x[512][32] ping-pong
  int*   bar   = (int*)(hbuf + 2 * BATCH * HID);        // [128] step barriers

  init_kernel<<<dim3(128), dim3(256), 0, stream>>>(hbuf, bar);

  embed_gates_kernel<<<dim3(G4 / 128, (SEQL * BATCH) / 64), dim3(256), 0,
                       stream>>>(x, emb, W_ih, b_ih, b_hh, xg);

  const size_t lstm_lds = (size_t)(HID * 128 + HID * BATCH) * sizeof(float);
  lstm_scan_kernel<<<dim3(LSTM_WGS), dim3(256), lstm_lds, stream>>>(
      xg, W_hh, hbuf, hsbuf, bar);

  fc_kernel<<<dim3(OUTD / 128, (SEQL * BATCH) / 64), dim3(256), 0, stream>>>(
      hsbuf, W_fc, b_fc, out);
}